// GraphProjection_38852274160230
// MI455X (gfx1250) — compile-verified
//
#include <hip/hip_runtime.h>
#include <stdint.h>

// ---------------------------------------------------------------------------
// GraphProjection for MI455X (gfx1250, wave32).
//
// Roofline: output = 100000 x 963 f32 = 385 MB streamed once -> ~16.5 us at
// 23.3 TB/s HBM. Feature pyramid is ~1.5 MB (L2-resident). Compute ~0.7 GFLOP.
// => pure write-bandwidth-bound gather kernel; WMMA does not apply (per-point
// gathered B operand would need a 94%-zero block-diagonal A).
//
// CDNA5-specific paths used:
//   * global_load_async_to_lds_b128 + s_wait_asynccnt: stage feat4
//     (7x7x512 f32 = 98 KB, covers 512/963 output cols) into the 320 KB WGP
//     LDS; gathered with ds_load_b128.
//   * non-temporal stores (th:TH_STORE_NT) for the 385 MB output stream so
//     it does not evict the hot feature maps from L2.
//   * wave32 mapping: one wave per point, lanes stride channels -> every
//     store iteration is a contiguous 128B+ wave transaction.
// ---------------------------------------------------------------------------

template <int C, int SIZE>
__device__ __forceinline__ void sample_level(
    float* __restrict__ op, const float* __restrict__ feat,
    float x, float y, int lane)
{
    // floor/ceil bilinear exactly as the reference (jnp.floor / jnp.ceil),
    // including the all-zero-weight quirk at integral coordinates.
    float x1 = floorf(x), x2 = ceilf(x);
    float y1 = floorf(y), y2 = ceilf(y);
    // Weights from UNCLAMPED coords (reference uses raw x1/x2/y1/y2)...
    float w11 = (x2 - x) * (y2 - y);
    float w21 = (x - x1) * (y2 - y);
    float w12 = (x2 - x) * (y - y1);
    float w22 = (x - x1) * (y - y1);
    // ...but JAX clamps out-of-range integer indices (ceil(223/4)=56 -> 55).
    constexpr int mx = SIZE - 1;
    int xi1 = min(max((int)x1, 0), mx);
    int xi2 = min(max((int)x2, 0), mx);
    int yi1 = min(max((int)y1, 0), mx);
    int yi2 = min(max((int)y2, 0), mx);

    const float* q11 = feat + (xi1 * SIZE + yi1) * C;
    const float* q12 = feat + (xi1 * SIZE + yi2) * C;
    const float* q21 = feat + (xi2 * SIZE + yi1) * C;
    const float* q22 = feat + (xi2 * SIZE + yi2) * C;

    // Reference summation order: w11*Q11 + w21*Q21 + w12*Q12 + w22*Q22.
    if constexpr (C >= 128) {
        // float4 gathers: corner bases are C-aligned (C % 4 == 0) and the
        // feature buffers are 16B aligned -> global_load_b128 / ds_load_b128.
        for (int v = lane; v < C / 4; v += 32) {
            float4 a = ((const float4*)q11)[v];
            float4 b = ((const float4*)q21)[v];
            float4 c4 = ((const float4*)q12)[v];
            float4 d = ((const float4*)q22)[v];
            int c = v * 4;
            __builtin_nontemporal_store(w11 * a.x + w21 * b.x + w12 * c4.x + w22 * d.x, op + c + 0);
            __builtin_nontemporal_store(w11 * a.y + w21 * b.y + w12 * c4.y + w22 * d.y, op + c + 1);
            __builtin_nontemporal_store(w11 * a.z + w21 * b.z + w12 * c4.z + w22 * d.z, op + c + 2);
            __builtin_nontemporal_store(w11 * a.w + w21 * b.w + w12 * c4.w + w22 * d.w, op + c + 3);
        }
    } else {
        // C == 64: float2 keeps all 32 lanes active in a single iteration.
        for (int v = lane; v < C / 2; v += 32) {
            float2 a = ((const float2*)q11)[v];
            float2 b = ((const float2*)q21)[v];
            float2 c2 = ((const float2*)q12)[v];
            float2 d = ((const float2*)q22)[v];
            int c = v * 2;
            __builtin_nontemporal_store(w11 * a.x + w21 * b.x + w12 * c2.x + w22 * d.x, op + c + 0);
            __builtin_nontemporal_store(w11 * a.y + w21 * b.y + w12 * c2.y + w22 * d.y, op + c + 1);
        }
    }
}

extern "C" __global__ __launch_bounds__(256)
void graph_projection_kernel(const float* __restrict__ coord,
                             const float* __restrict__ feat1,
                             const float* __restrict__ feat2,
                             const float* __restrict__ feat3,
                             const float* __restrict__ feat4,
                             float* __restrict__ out,
                             int N)
{
    // Dynamic LDS: 7*7*512 floats = 100,352 bytes (3 workgroups / 320KB WGP).
    extern __shared__ float s_feat4[];
    constexpr int F4_ELTS  = 7 * 7 * 512;    // 25088 floats
    constexpr int F4_VEC16 = F4_ELTS / 4;    // 6272 x 16B transfers (%32==0)

    // ---- Stage feat4 into LDS with CDNA5 async global->LDS DMA ------------
    {
        // Low 32 bits of a flat LDS pointer ARE the LDS byte offset
        // (addrspacecast local->flat keeps the offset in addr[31:0]).
        const unsigned lds_base = (unsigned)(uintptr_t)s_feat4;
        const uint64_t gbase    = (uint64_t)(uintptr_t)feat4;
        for (int i = threadIdx.x; i < F4_VEC16; i += 256) {
            unsigned off      = (unsigned)i * 16u;
            unsigned lds_addr = lds_base + off;
            // GVS mode: mem = SGPR64 + VGPR32 offset; LDS dest = VGPR (vdst)
            asm volatile("global_load_async_to_lds_b128 %0, %1, %2"
                         :
                         : "v"(lds_addr), "v"(off), "s"(gbase)
                         : "memory");
        }
        asm volatile("s_wait_asynccnt 0" ::: "memory");
        __syncthreads();
    }

    // ---- One wave32 per point; lanes stride across channels ---------------
    const int lane          = threadIdx.x & 31;
    const int wave_in_block = threadIdx.x >> 5;
    const int waves_per_blk = blockDim.x >> 5;
    const int wave_id       = blockIdx.x * waves_per_blk + wave_in_block;
    const int n_waves       = gridDim.x * waves_per_blk;

    for (int n = wave_id; n < N; n += n_waves) {
        // n is wave-uniform; tell the compiler so coord loads scalarize
        // to SMEM and row-pointer math goes to SALU.
        const int un = __builtin_amdgcn_readfirstlane(n);

        float X = coord[un * 3 + 0];
        float Y = coord[un * 3 + 1];
        float Z = coord[un * 3 + 2];

        float h = 250.0f * (-Y) / (-Z) + 112.0f;
        float w = 250.0f * X / (-Z) + 112.0f;
        h = fminf(fmaxf(h, 0.0f), 223.0f);
        w = fminf(fmaxf(w, 0.0f), 223.0f);

        float* op = out + (size_t)un * 963;

        if (lane == 0) {
            __builtin_nontemporal_store(X, op + 0);
            __builtin_nontemporal_store(Y, op + 1);
            __builtin_nontemporal_store(Z, op + 2);
        }

        // scales 224/size = 4, 8, 16, 32 -> exact reciprocal multiplies
        sample_level< 64, 56>(op + 3,   feat1,   h * 0.25f,    w * 0.25f,    lane);
        sample_level<128, 28>(op + 67,  feat2,   h * 0.125f,   w * 0.125f,   lane);
        sample_level<256, 14>(op + 195, feat3,   h * 0.0625f,  w * 0.0625f,  lane);
        sample_level<512,  7>(op + 451, s_feat4, h * 0.03125f, w * 0.03125f, lane);
    }
}

extern "C" void kernel_launch(void* const* d_in, const int* in_sizes, int n_in,
                              void* d_out, int out_size, void* d_ws, size_t ws_size,
                              hipStream_t stream) {
    const float* coord = (const float*)d_in[0];
    const float* feat1 = (const float*)d_in[1];
    const float* feat2 = (const float*)d_in[2];
    const float* feat3 = (const float*)d_in[3];
    const float* feat4 = (const float*)d_in[4];
    float* out = (float*)d_out;

    const int N = in_sizes[0] / 3;

    const dim3 grid(512);
    const dim3 block(256);
    const size_t shmem = (size_t)(7 * 7 * 512) * sizeof(float);  // 98 KB

    graph_projection_kernel<<<grid, block, shmem, stream>>>(
        coord, feat1, feat2, feat3, feat4, out, N);
}